// EdgeSIMPLEBatched_39298950758997
// MI455X (gfx1250) — compile-verified
//
#include <hip/hip_runtime.h>
#include <hip/hip_bf16.h>
#include <stdint.h>

// SIMPLE k-subset layer for (128, 4096, 2) fp32 scores, k=32, T=1.
// One wave32 workgroup per BE-row (256 rows). Lane l holds esp degree j=l+1
// (e_0 == 1 -> log 0 handled implicitly), so the k+1=33 wide log-esp scan maps
// exactly onto wave32. Backward sweep stores checkpoints every 128 steps in
// LDS; each forward block recomputes its B tile into an LDS ring (checkpoint
// recompute keeps everything inside the 320KB WGP LDS -- no HBM intermediate).
// Theta row is streamed into LDS via the CDNA5 async global->LDS DMA path.

#define NEGC   (-1.0e30f)
#define NROWS  256
#define NN     4096
#define KK     32
#define BLK    128
#define NBLK   32
#define HALFC  524288u      // threefry counter split: (N*T*BE)/2

__device__ __forceinline__ float lse2(float a, float b) {
    float m = fmaxf(a, b);
    float d = fabsf(a - b);
    return m + __logf(1.0f + __expf(-d));
}

__device__ __forceinline__ void threefry2x32(unsigned k0, unsigned k1,
                                             unsigned& x0, unsigned& x1) {
    const unsigned ks2 = k0 ^ k1 ^ 0x1BD11BDAu;
    x0 += k0; x1 += k1;
#define TF_R(r) { x0 += x1; x1 = (x1 << (r)) | (x1 >> (32 - (r))); x1 ^= x0; }
    TF_R(13) TF_R(15) TF_R(26) TF_R(6)
    x0 += k1;  x1 += ks2 + 1u;
    TF_R(17) TF_R(29) TF_R(16) TF_R(24)
    x0 += ks2; x1 += k0 + 2u;
    TF_R(13) TF_R(15) TF_R(26) TF_R(6)
    x0 += k0;  x1 += k1 + 3u;
    TF_R(17) TF_R(29) TF_R(16) TF_R(24)
    x0 += k1;  x1 += ks2 + 4u;
    TF_R(13) TF_R(15) TF_R(26) TF_R(6)
    x0 += ks2; x1 += k0 + 5u;
#undef TF_R
}

__device__ __forceinline__ float u01(unsigned b) {
    return __uint_as_float((b >> 9) | 0x3f800000u) - 1.0f;
}

__global__ __launch_bounds__(32)
void simple_subset_kernel(const float* __restrict__ scores, float* __restrict__ out) {
    const int be   = blockIdx.x;        // 0..255   (row = batch*2 + ensemble)
    const int bb   = be >> 1;
    const int ee   = be & 1;
    const int lane = threadIdx.x;       // 0..31

    __shared__ float thetaLDS[NN];          // 16 KB
    __shared__ float Bring[(BLK + 1) * 32]; // 16.5 KB : B[i] tile for current block
    __shared__ float chk[(NBLK - 1) * 32];  // ~4 KB   : B[(b+1)*128] checkpoints
    __shared__ float uall[NN];              // 16 KB   : threefry uniforms
    __shared__ float stg[2 * BLK];          // 1 KB    : output staging

    const float* rowbase = scores + (size_t)bb * NN * 2 + ee;

    // ---- stream theta row into LDS via CDNA5 async global->LDS DMA ----
    for (int t0 = 0; t0 < NN; t0 += 32) {
        unsigned lds = (unsigned)(uintptr_t)(void*)&thetaLDS[t0 + lane];
        uint64_t ga  = (uint64_t)(uintptr_t)(rowbase + (size_t)(t0 + lane) * 2);
        asm volatile("global_load_async_to_lds_b32 %0, %1, off"
                     :: "v"(lds), "v"(ga) : "memory");
    }
    asm volatile("s_wait_asynccnt 0" ::: "memory");
    __syncthreads();   // single-wave WG: degenerates to a NOP/wait

    // ================= Phase A: backward esp sweep, store checkpoints ======
    // After processing index i, state S == B[i][lane+1] (esp of theta[i..N-1]).
    float S = NEGC;
    #pragma unroll 4
    for (int i = NN - 1; i >= 0; --i) {
        float th   = thetaLDS[i];
        float prev = __shfl_up(S, 1, 32);
        if (lane == 0) prev = 0.0f;        // log e_0 = 0
        S = lse2(S, prev + th);
        if ((i & (BLK - 1)) == 0 && i > 0)
            chk[((i >> 7) - 1) * 32 + lane] = S;
    }
    const float logZ = __shfl(S, 31, 32);  // F[N][k] == B[0][32]

    // ================= Phase B: blockwise recompute + forward/sample =======
    int   r = KK;       // conditional-Poisson remaining count (wave-uniform)
    float F = NEGC;     // F[i][lane+1]

    for (int b = 0; b < NBLK; ++b) {
        const int s = b << 7;

        // --- B-ring recompute for this block: Bring[m] = B[s+m][lane+1] ---
        float Sb = (b == NBLK - 1) ? NEGC : chk[b * 32 + lane];
        Bring[BLK * 32 + lane] = Sb;
        #pragma unroll 4
        for (int m = BLK - 1; m >= 0; --m) {
            float th   = thetaLDS[s + m];
            float prev = __shfl_up(Sb, 1, 32);
            if (lane == 0) prev = 0.0f;
            Sb = lse2(Sb, prev + th);
            Bring[m * 32 + lane] = Sb;
        }

        // --- threefry uniforms: counter pair (idx, idx+HALFC); the partner
        //     output belongs to position i+2048 of the SAME row -> stash it.
        if (b < 16) {
            #pragma unroll
            for (int q = 0; q < 4; ++q) {
                int ip = s + q * 32 + lane;                // i' in [0,2048)
                unsigned c0 = (unsigned)(ip * NROWS + be);
                unsigned c1 = c0 + HALFC;
                threefry2x32(0u, 42u, c0, c1);
                uall[ip]        = u01(c0);
                uall[ip + 2048] = u01(c1);
            }
        }

        // --- fused forward sweep: marginals + conditional-Poisson sampling --
        for (int m = 0; m < BLK; ++m) {
            const int i = s + m;
            float th = thetaLDS[i];
            float Bn = Bring[(m + 1) * 32 + lane];          // B[i+1][lane+1]

            float Fv = __shfl_up(F, 1, 32);                 // F[i][lane]
            if (lane == 0) Fv = 0.0f;
            float Bv = (lane == 31) ? 0.0f
                                    : __shfl(Bn, 30 - lane, 32); // B[i+1][31-lane]
            float t = Fv + Bv;                              // comb term, j=lane

            // wave32 logsumexp over the 32 comb terms
            float M = t;
            #pragma unroll
            for (int o = 16; o; o >>= 1) M = fmaxf(M, __shfl_xor(M, o, 32));
            float ex = __expf(t - M);
            #pragma unroll
            for (int o = 16; o; o >>= 1) ex += __shfl_xor(ex, o, 32);
            float red  = M + __logf(ex);                    // log e_{k-1}(w \ i)
            float marg = __expf(th + red - logZ);

            // sampling step (all quantities wave-uniform)
            float u   = uall[i];
            int   rm1 = r - 1;
            float Bn_r = (rm1 <= 0) ? 0.0f : __shfl(Bn, rm1 - 1, 32); // B[i+1][r-1]
            float Bc_r = (r == 0)   ? 0.0f : Bring[m * 32 + (r - 1)]; // B[i][r]
            float logq = th + Bn_r - Bc_r;
            int incl = (r > 0) && (u < __expf(fminf(logq, 0.0f)));
            float sv = incl ? 1.0f : 0.0f;
            sv = (sv - marg) + marg;   // faithful to straight-through forward
            r -= incl;

            // F scan update: F[j] <- logaddexp(F[j], F[j-1] + th)
            F = lse2(F, Fv + th);

            if (lane == 0) { stg[m] = marg; stg[BLK + m] = sv; }
        }

        // --- coalesced flush of this block's outputs (stride-2 in ens dim) --
        const size_t obase = ((size_t)bb * NN + s) * 2 + ee;
        #pragma unroll
        for (int q = 0; q < 4; ++q) {
            int m = q * 32 + lane;
            out[obase + (size_t)m * 2]                          = stg[BLK + m]; // mask
            out[(size_t)NROWS * NN * 2 + obase + (size_t)m * 2] = stg[m];       // marginals
        }
    }
}

extern "C" void kernel_launch(void* const* d_in, const int* in_sizes, int n_in,
                              void* d_out, int out_size, void* d_ws, size_t ws_size,
                              hipStream_t stream) {
    (void)in_sizes; (void)n_in; (void)out_size; (void)d_ws; (void)ws_size;
    const float* scores = (const float*)d_in[0];
    float* out = (float*)d_out;
    simple_subset_kernel<<<dim3(NROWS), dim3(32), 0, stream>>>(scores, out);
}